// QuantumNeuralDecoder_9646496547587
// MI455X (gfx1250) — compile-verified
//
#include <hip/hip_runtime.h>
#include <hip/hip_bf16.h>

// ---------------------------------------------------------------------------
// QuantumNeuralDecoder on MI455X (gfx1250)
//   B=256 samples, NQ=16 qubits -> 65536 complex64 amps/sample (128MB total,
//   fits in 192MB L2). Strategy:
//     encode  : WMMA f32 16x16x4 GEMM -> angles -> (cos,sin) table
//     init    : product-state |psi> = RY(angles)|0..0>  (analytic, one pass)
//     layer l : half_kernel  - qubits 1..15 applied inside 256KB LDS
//               cross_kernel - qubit 0 gate + diagonal ZZ phase, streaming
//               (single-qubit gates on distinct qubits commute -> reorder ok)
//     final   : Z expectations fused into last cross pass (wave32 shfl reduce)
//     decode  : tiny MLP
// ---------------------------------------------------------------------------

typedef __attribute__((ext_vector_type(2))) float v2f;
typedef __attribute__((ext_vector_type(8))) float v8f;
typedef __attribute__((ext_vector_type(4))) unsigned int v4u;
typedef __attribute__((ext_vector_type(8))) int v8i;
typedef __attribute__((ext_vector_type(4))) int v4i;

#if defined(__has_builtin)
#if __has_builtin(__builtin_amdgcn_tensor_load_to_lds) && \
    __has_builtin(__builtin_amdgcn_s_wait_tensorcnt)
#define TDM_OK 1
#endif
#endif
#ifndef TDM_OK
#define TDM_OK 0
#endif

#define NQ 16
#define NSTATES 65536
#define HALF 32768
#define PI_F 3.14159265358979323846f

__device__ __forceinline__ float2 cmul(float2 a, float2 b) {
    return make_float2(a.x * b.x - a.y * b.y, a.x * b.y + a.y * b.x);
}
__device__ __forceinline__ float2 cadd(float2 a, float2 b) {
    return make_float2(a.x + b.x, a.y + b.y);
}

struct U2 { float2 u00, u01, u10, u11; };

// U = RX(t0) @ RY(t1) @ RZ(t2)
__device__ __forceinline__ U2 make_gate(float t0, float t1, float t2) {
    float c0 = cosf(0.5f * t0), s0 = sinf(0.5f * t0);
    float c1 = cosf(0.5f * t1), s1 = sinf(0.5f * t1);
    float cz = cosf(0.5f * t2), sz = sinf(0.5f * t2);
    float2 m00 = make_float2(c0 * c1, -s0 * s1);
    float2 m01 = make_float2(-c0 * s1, -s0 * c1);
    float2 m10 = make_float2(c0 * s1, -s0 * c1);
    float2 m11 = make_float2(c0 * c1, s0 * s1);
    float2 em = make_float2(cz, -sz), ep = make_float2(cz, sz);
    U2 u;
    u.u00 = cmul(m00, em); u.u01 = cmul(m01, ep);
    u.u10 = cmul(m10, em); u.u11 = cmul(m11, ep);
    return u;
}

// ring-ZZ phase factor for basis state i (MSB-first qubit order)
__device__ __forceinline__ float2 phase_of(int i, const float* ph) {
    int u = (i ^ ((i << 1) | (i >> 15))) & 0xFFFF; // bit(15-q) = b_q ^ b_{q+1 mod 16}
    float dot = 0.f;
#pragma unroll
    for (int q = 0; q < NQ; ++q)
        dot += ((u >> (15 - q)) & 1) ? -ph[q] : ph[q];
    float ang = -0.5f * dot;
    return make_float2(cosf(ang), sinf(ang));
}

// ---------------------------------------------------------------------------
// 1) angles = tanh(x @ W_enc + b) * pi  via v_wmma_f32_16x16x4_f32
//    grid 16 x 32 threads (one wave per 16-row tile); writes (cos,sin)(ang/2)
// ---------------------------------------------------------------------------
__global__ __launch_bounds__(32) void encode_kernel(
    const float* __restrict__ x, const float* __restrict__ W,
    const float* __restrict__ b, float2* __restrict__ cs) {
    const int lane = threadIdx.x;
    const int m0 = blockIdx.x * 16;
    const int mr = lane & 15;       // A: M row / B: N col
    const int kh = lane >> 4;       // K half-select (K, K+2)
    v8f acc = {};
    for (int k0 = 0; k0 < 512; k0 += 4) {
        v2f a, bb;
        const float* xp = x + (size_t)(m0 + mr) * 512 + k0 + 2 * kh;
        a[0] = xp[0]; a[1] = xp[1];
        const float* wp = W + (size_t)(k0 + 2 * kh) * 16 + mr;
        bb[0] = wp[0]; bb[1] = wp[16];
        acc = __builtin_amdgcn_wmma_f32_16x16x4_f32(
            false, a, false, bb, (short)0, acc, false, false);
    }
#pragma unroll
    for (int r = 0; r < 8; ++r) {
        int mm = m0 + r + 8 * kh;   // D row: lanes 0-15 -> M=r, 16-31 -> M=r+8
        float ang = tanhf(acc[r] + b[mr]) * PI_F;
        cs[mm * NQ + mr] = make_float2(cosf(0.5f * ang), sinf(0.5f * ang));
    }
}

// ---------------------------------------------------------------------------
// 2) product state: amp(i) = prodHi(i>>8) * prodLo(i&255), imag = 0
// ---------------------------------------------------------------------------
__global__ __launch_bounds__(256) void init_kernel(
    const float2* __restrict__ cs, float2* __restrict__ state) {
    __shared__ float pHi[256], pLo[256];
    const int t = threadIdx.x, s = blockIdx.x;
    float ph = 1.f, pl = 1.f;
#pragma unroll
    for (int q = 0; q < 8; ++q) {
        float2 c = cs[s * NQ + q];
        ph *= ((t >> (7 - q)) & 1) ? c.y : c.x;
    }
#pragma unroll
    for (int q = 8; q < 16; ++q) {
        float2 c = cs[s * NQ + q];
        pl *= ((t >> (15 - q)) & 1) ? c.y : c.x;
    }
    pHi[t] = ph; pLo[t] = pl;
    __syncthreads();
    float2* st = state + (size_t)s * NSTATES;
    for (int j = 0; j < 256; ++j)
        st[j * 256 + t] = make_float2(pHi[j] * pLo[t], 0.f); // coalesced over t
}

// ---------------------------------------------------------------------------
// 3) half kernel: load 32768-amp half (qubit-0 bit fixed) into 256KB LDS via
//    TDM, apply gates on qubits 1..15, store back. grid = 512 (sample*2+h).
// ---------------------------------------------------------------------------
__global__ __launch_bounds__(1024) void half_kernel(
    float2* __restrict__ state, const float* __restrict__ theta, int layer) {
    extern __shared__ float2 S[];
    const int tid = threadIdx.x;
    const int s = blockIdx.x >> 1, h = blockIdx.x & 1;
    float2* gst = state + (size_t)s * NSTATES + (size_t)h * HALF;

#if TDM_OK
    if (tid == 0) {
        unsigned long long ga = (unsigned long long)(size_t)gst;
        unsigned int lds = (unsigned int)(size_t)(&S[0]); // low 32b = LDS addr
        v4u g0;
        g0[0] = 1u;                                  // count=1 valid descriptor
        g0[1] = lds;                                 // lds_addr
        g0[2] = (unsigned int)ga;                    // global_addr[31:0]
        g0[3] = (unsigned int)((ga >> 32) & 0x1FFFFFFull) | 0x80000000u; // +type=2
        v8i g1;
        g1[0] = 0x00030000;       // data_size=3 (8B), mask=0, no pad/iter
        g1[1] = (int)0x80000000u; // tensor_dim0 = 32768  ([31:16] of w1)
        g1[2] = (int)0x00010000u; // tensor_dim0 hi=0, tensor_dim1 = 1
        g1[3] = (int)0x80000000u; // tile_dim0 = 32768
        g1[4] = 1;                // tile_dim1 = 1, tile_dim2 = 0
        g1[5] = HALF;             // tensor_dim0_stride lo
        g1[6] = 0; g1[7] = 0;
        v4i gz4 = {0, 0, 0, 0};
        v8i gz8 = {0, 0, 0, 0, 0, 0, 0, 0};
        __builtin_amdgcn_tensor_load_to_lds(g0, g1, gz4, gz4, gz8, 0);
        __builtin_amdgcn_s_wait_tensorcnt(0);
    }
#else
    {
        const float4* g4 = (const float4*)gst;
        float4* s4 = (float4*)S;
        for (int i = tid; i < HALF / 2; i += 1024) s4[i] = g4[i];
    }
#endif
    __syncthreads();

#pragma unroll 1
    for (int q = 1; q < NQ; ++q) {
        const float* tp = theta + (size_t)(layer * NQ + q) * 3;
        U2 u = make_gate(tp[0], tp[1], tp[2]);
        const int stride = 1 << (15 - q); // pair stride within 15-bit local idx
#pragma unroll
        for (int pp = 0; pp < 16; ++pp) {
            int p = tid + pp * 1024;
            int i0 = ((p & ~(stride - 1)) << 1) | (p & (stride - 1));
            int i1 = i0 + stride;
            float2 a0 = S[i0], a1 = S[i1];
            float2 n0 = cadd(cmul(u.u00, a0), cmul(u.u01, a1));
            float2 n1 = cadd(cmul(u.u10, a0), cmul(u.u11, a1));
            S[i0] = n0; S[i1] = n1;
        }
        __syncthreads();
    }

    {   // store back (coalesced b128)
        const float4* s4 = (const float4*)S;
        float4* g4 = (float4*)gst;
        for (int i = tid; i < HALF / 2; i += 1024) g4[i] = s4[i];
    }
}

// ---------------------------------------------------------------------------
// 4) cross kernel: qubit-0 gate across halves + diagonal ZZ phase.
//    final layer: fuse |amp|^2 -> <Z_q> reduction (saves a 128MB pass).
// ---------------------------------------------------------------------------
__global__ __launch_bounds__(1024) void cross_kernel(
    float2* __restrict__ state, const float* __restrict__ theta,
    const float* __restrict__ phi, float* __restrict__ zout,
    int layer, int final) {
    const int tid = threadIdx.x, s = blockIdx.x;
    float2* st = state + (size_t)s * NSTATES;
    const float* tp = theta + (size_t)(layer * NQ + 0) * 3;
    U2 u = make_gate(tp[0], tp[1], tp[2]);
    float ph[NQ];
#pragma unroll
    for (int q = 0; q < NQ; ++q) ph[q] = phi[layer * NQ + q];
    float zacc[NQ];
#pragma unroll
    for (int q = 0; q < NQ; ++q) zacc[q] = 0.f;

    for (int it = 0; it < 32; ++it) {
        int i0 = tid + it * 1024;     // qubit-0 bit = 0 side
        int i1 = i0 + HALF;           // qubit-0 bit = 1 side
        float2 a0 = st[i0], a1 = st[i1];
        float2 n0 = cadd(cmul(u.u00, a0), cmul(u.u01, a1));
        float2 n1 = cadd(cmul(u.u10, a0), cmul(u.u11, a1));
        n0 = cmul(n0, phase_of(i0, ph));
        n1 = cmul(n1, phase_of(i1, ph));
        if (!final) {
            st[i0] = n0; st[i1] = n1;
        } else {
            float p0 = n0.x * n0.x + n0.y * n0.y;
            float p1 = n1.x * n1.x + n1.y * n1.y;
#pragma unroll
            for (int q = 0; q < NQ; ++q) {
                zacc[q] += (((i0 >> (15 - q)) & 1) ? -p0 : p0)
                         + (((i1 >> (15 - q)) & 1) ? -p1 : p1);
            }
        }
    }

    if (final) {
        __shared__ float red[32][NQ];
#pragma unroll
        for (int q = 0; q < NQ; ++q) {
            float v = zacc[q];
            for (int off = 16; off > 0; off >>= 1) v += __shfl_xor(v, off, 32);
            zacc[q] = v;
        }
        int wave = tid >> 5, lane = tid & 31;
        if (lane == 0) {
#pragma unroll
            for (int q = 0; q < NQ; ++q) red[wave][q] = zacc[q];
        }
        __syncthreads();
        if (tid < NQ) {
            float v = 0.f;
            for (int w = 0; w < 32; ++w) v += red[w][tid];
            zout[s * NQ + tid] = v;
        }
    }
}

// ---------------------------------------------------------------------------
// 5) decoder MLP: out = relu(z@W1+b1)@W2 + b2   (256x16 -> 32 -> 10)
// ---------------------------------------------------------------------------
__global__ __launch_bounds__(256) void decode_kernel(
    const float* __restrict__ z, const float* __restrict__ W1,
    const float* __restrict__ b1, const float* __restrict__ W2,
    const float* __restrict__ b2, float* __restrict__ out) {
    __shared__ float sW1[16 * 32], sB1[32], sW2[32 * 10], sB2[10];
    const int t = threadIdx.x;
    for (int i = t; i < 16 * 32; i += 256) sW1[i] = W1[i];
    for (int i = t; i < 32 * 10; i += 256) sW2[i] = W2[i];
    if (t < 32) sB1[t] = b1[t];
    if (t < 10) sB2[t] = b2[t];
    __syncthreads();
    float zz[NQ];
#pragma unroll
    for (int q = 0; q < NQ; ++q) zz[q] = z[t * NQ + q];
    float acc[10];
#pragma unroll
    for (int o = 0; o < 10; ++o) acc[o] = sB2[o];
    for (int j = 0; j < 32; ++j) {
        float hv = sB1[j];
#pragma unroll
        for (int q = 0; q < NQ; ++q) hv = fmaf(zz[q], sW1[q * 32 + j], hv);
        hv = fmaxf(hv, 0.f);
#pragma unroll
        for (int o = 0; o < 10; ++o) acc[o] = fmaf(hv, sW2[j * 10 + o], acc[o]);
    }
#pragma unroll
    for (int o = 0; o < 10; ++o) out[t * 10 + o] = acc[o];
}

// ---------------------------------------------------------------------------
extern "C" void kernel_launch(void* const* d_in, const int* in_sizes, int n_in,
                              void* d_out, int out_size, void* d_ws, size_t ws_size,
                              hipStream_t stream) {
    const float* x     = (const float*)d_in[0];
    const float* W_enc = (const float*)d_in[1];
    const float* b_enc = (const float*)d_in[2];
    const float* theta = (const float*)d_in[3];
    const float* phi   = (const float*)d_in[4];
    const float* W1    = (const float*)d_in[5];
    const float* b1    = (const float*)d_in[6];
    const float* W2    = (const float*)d_in[7];
    const float* b2    = (const float*)d_in[8];
    float* out = (float*)d_out;

    char* ws = (char*)d_ws;
    float2* cs    = (float2*)ws;                 //  32 KB: (cos,sin)(ang/2)
    float*  zbuf  = (float*)(ws + 8192);         //  16 KB: <Z_q>
    float2* state = (float2*)(ws + 32768);       // 128 MB: statevectors

    encode_kernel<<<16, 32, 0, stream>>>(x, W_enc, b_enc, cs);
    init_kernel<<<256, 256, 0, stream>>>(cs, state);
    for (int l = 0; l < 4; ++l) {
        half_kernel<<<512, 1024, 262144, stream>>>(state, theta, l);
        cross_kernel<<<256, 1024, 0, stream>>>(state, theta, phi, zbuf, l, l == 3);
    }
    decode_kernel<<<1, 256, 0, stream>>>(zbuf, W1, b1, W2, b2, out);
}